// CategoryInformedGNNLayer_88794153877954
// MI455X (gfx1250) — compile-verified
//
#include <hip/hip_runtime.h>
#include <stdint.h>

typedef float v2f __attribute__((ext_vector_type(2)));
typedef float v8f __attribute__((ext_vector_type(8)));

#define NN 100000
#define NE 3200000
#define D  128
#define MT (NN / 16)   // 6250 M-tiles, exact

// --- 1) deg[i] = 1.0 (self-loop weight) -------------------------------------
__global__ void k_deg_init(float* __restrict__ deg) {
    int i = blockIdx.x * blockDim.x + threadIdx.x;
    if (i < NN) deg[i] = 1.0f;
}

// --- 2) deg[dst[e]] += w[e] --------------------------------------------------
__global__ void k_deg_acc(const long long* __restrict__ dst,
                          const float* __restrict__ w,
                          float* __restrict__ deg) {
    int e = blockIdx.x * blockDim.x + threadIdx.x;
    if (e < NE) unsafeAtomicAdd(&deg[(size_t)dst[e]], w[e]);
}

// --- 3) deg -> deg^{-1/2} in place ------------------------------------------
__global__ void k_rsqrt(float* __restrict__ deg) {
    int i = blockIdx.x * blockDim.x + threadIdx.x;
    if (i < NN) {
        float d = deg[i];
        deg[i] = d > 0.0f ? rsqrtf(d) : 0.0f;
    }
}

// --- 4) h = x @ W via V_WMMA_F32_16X16X4_F32 --------------------------------
// 8 waves/block; wave handles one 16-row M-tile across all 8 N-tiles (A reuse).
// W staged in LDS (64KB of the 320KB WGP pool).
__global__ void __launch_bounds__(256) k_gemm(const float* __restrict__ x,
                                              const float* __restrict__ W,
                                              float* __restrict__ h) {
    __shared__ float Wlds[D * D];  // 64 KB
    for (int i = threadIdx.x; i < (D * D) / 4; i += 256)
        ((float4*)Wlds)[i] = ((const float4*)W)[i];
    __syncthreads();

    const int wave = threadIdx.x >> 5;
    const int lane = threadIdx.x & 31;
    const int half = lane >> 4;   // 0: K=0,1 ; 1: K=2,3 (A/B frag halves)
    const int l    = lane & 15;

    const int tileM = blockIdx.x * 8 + wave;   // wave-uniform -> EXEC stays all-1s
    if (tileM >= MT) return;

    const float* __restrict__ xrow = x + (size_t)(tileM * 16 + l) * D;

    v8f acc[8];
#pragma unroll
    for (int n = 0; n < 8; ++n) acc[n] = (v8f){0, 0, 0, 0, 0, 0, 0, 0};

    for (int k = 0; k < D; k += 4) {
        // A 16x4 frag: lanes 0-15 K=k..k+1, lanes 16-31 K=k+2..k+3 (b64 load)
        v2f a = *(const v2f*)(xrow + k + 2 * half);
#pragma unroll
        for (int n = 0; n < 8; ++n) {
            const int col = n * 16 + l;
            v2f b;
            b.x = Wlds[(k + 2 * half) * D + col];
            b.y = Wlds[(k + 2 * half + 1) * D + col];
            acc[n] = __builtin_amdgcn_wmma_f32_16x16x4_f32(
                false, a, false, b, (short)0, acc[n], false, false);
        }
    }

    // C/D layout: VGPR j -> row (j + 8*half), col = lane&15 (coalesced 64B rows)
#pragma unroll
    for (int n = 0; n < 8; ++n) {
        float* hp = h + (size_t)(tileM * 16 + 8 * half) * D + n * 16 + l;
#pragma unroll
        for (int j = 0; j < 8; ++j) hp[(size_t)j * D] = acc[n][j];
    }
}

// --- 5) out = b + h * dis^2 (bias + self-loop message) ----------------------
__global__ void k_out_init(const float* __restrict__ h,
                           const float* __restrict__ dis,
                           const float* __restrict__ b,
                           float* __restrict__ out) {
    long long idx = (long long)blockIdx.x * blockDim.x + threadIdx.x;
    if (idx < (long long)NN * D) {
        int n = (int)(idx >> 7);
        int c = (int)(idx & (D - 1));
        float di = dis[n];
        out[idx] = b[c] + h[idx] * di * di;
    }
}

// --- 6) out[dst] += h[src] * (dis[src]*w*dis[dst]) --------------------------
// One wave per edge: lane i handles 4 consecutive floats (float4 gather,
// 4x global_atomic_add_f32). h/out live in L2 (51MB each vs 192MB L2).
__global__ void k_scatter(const long long* __restrict__ src,
                          const long long* __restrict__ dst,
                          const float* __restrict__ w,
                          const float* __restrict__ dis,
                          const float* __restrict__ h,
                          float* __restrict__ out) {
    long long t = (long long)blockIdx.x * blockDim.x + threadIdx.x;
    int e    = (int)(t >> 5);
    int lane = (int)(t & 31);
    if (e < NE) {
        if (e + 2048 < NE) __builtin_prefetch(&src[e + 2048], 0, 0);
        long long s = src[e];
        long long d = dst[e];
        float nrm = dis[(size_t)s] * w[e] * dis[(size_t)d];
        float4 hv = *(const float4*)(h + (size_t)s * D + lane * 4);
        float* op = out + (size_t)d * D + lane * 4;
        unsafeAtomicAdd(op + 0, hv.x * nrm);
        unsafeAtomicAdd(op + 1, hv.y * nrm);
        unsafeAtomicAdd(op + 2, hv.z * nrm);
        unsafeAtomicAdd(op + 3, hv.w * nrm);
    }
}

extern "C" void kernel_launch(void* const* d_in, const int* in_sizes, int n_in,
                              void* d_out, int out_size, void* d_ws, size_t ws_size,
                              hipStream_t stream) {
    const float*     x  = (const float*)d_in[0];
    const long long* ei = (const long long*)d_in[1];   // int64 [2, NE]
    const float*     w  = (const float*)d_in[2];
    const float*     W  = (const float*)d_in[3];
    const float*     b  = (const float*)d_in[4];

    const long long* src = ei;
    const long long* dst = ei + NE;

    float* out = (float*)d_out;

    // workspace: deg/dis [NN floats] then h [NN*D floats]
    float* deg = (float*)d_ws;
    size_t off = (((size_t)NN * sizeof(float)) + 1023) & ~(size_t)1023;
    float* h   = (float*)((char*)d_ws + off);

    k_deg_init<<<(NN + 255) / 256, 256, 0, stream>>>(deg);
    k_deg_acc <<<(NE + 255) / 256, 256, 0, stream>>>(dst, w, deg);
    k_rsqrt   <<<(NN + 255) / 256, 256, 0, stream>>>(deg);
    k_gemm    <<<(MT + 7) / 8,     256, 0, stream>>>(x, W, h);
    k_out_init<<<(int)(((long long)NN * D + 255) / 256), 256, 0, stream>>>(h, deg, b, out);
    k_scatter <<<(int)(((long long)NE * 32 + 255) / 256), 256, 0, stream>>>(src, dst, w, deg, h, out);
}